// NodeAELoss_50105088475385
// MI455X (gfx1250) — compile-verified
//
#include <hip/hip_runtime.h>
#include <math.h>

#define PMAX 128   // max persons supported (reference uses 64)
#define REP  4     // LDS sub-banking replication to reduce ds_add_f32 conflicts

typedef float v2f __attribute__((ext_vector_type(2)));
typedef float v8f __attribute__((ext_vector_type(8)));

// ---------------------------------------------------------------------------
// Kernel 0: zero the per-segment scratch (counts/sums/sumsq) and maxlabel.
// Must run every call: atomics accumulate and the harness doesn't re-poison.
// ---------------------------------------------------------------------------
__global__ void ae_zero_ws(float* __restrict__ acc, int nFloats,
                           int* __restrict__ maxl, int B) {
  int i = blockIdx.x * blockDim.x + threadIdx.x;
  int stride = gridDim.x * blockDim.x;
  for (int k = i; k < nFloats; k += stride) acc[k] = 0.0f;
  for (int k = i; k < B; k += stride) maxl[k] = -1;
}

// ---------------------------------------------------------------------------
// Kernel 1: single streaming pass. Each block owns a contiguous chunk of ONE
// image, accumulates (count, sum, sum^2) per person into LDS with 4-way
// replicated slots (ds_add_f32), then flushes once to global scratch with
// global_atomic_add_f32. batch_index is implicit (block -> image mapping).
// ---------------------------------------------------------------------------
__global__ __launch_bounds__(256) void ae_accum(
    const float* __restrict__ tags, const int* __restrict__ labels,
    const int* __restrict__ pP,
    float* __restrict__ counts, float* __restrict__ sums,
    float* __restrict__ sumsq, int* __restrict__ maxlab,
    int NPI, int bpi, int perBlock)
{
  __shared__ float s_cnt[PMAX * REP];
  __shared__ float s_sum[PMAX * REP];
  __shared__ float s_sq [PMAX * REP];
  __shared__ int   s_maxl;

  int P = *pP; if (P > PMAX) P = PMAX;
  const int tid = threadIdx.x;
  const int b = blockIdx.x / bpi;
  const int j = blockIdx.x - b * bpi;

  for (int p = tid; p < P * REP; p += blockDim.x) {
    s_cnt[p] = 0.0f; s_sum[p] = 0.0f; s_sq[p] = 0.0f;
  }
  if (tid == 0) s_maxl = -1;
  __syncthreads();

  const long base = (long)b * (long)NPI;
  int start = j * perBlock;
  int end = start + perBlock; if (end > NPI) end = NPI;
  int lmax = -1;
  const int sub = tid & (REP - 1);

  if (start < end) {
    const int n4 = (end - start) & ~3;
    // vectorized main loop: b128 tag + b128 label loads per thread
    for (int i = start + tid * 4; i < start + n4; i += blockDim.x * 4) {
      const float4 t4 = *(const float4*)(tags + base + i);
      const int4   l4 = *(const int4*)(labels + base + i);
      {
        int l = min(l4.x, PMAX - 1); lmax = max(lmax, l); int s = l * REP + sub;
        atomicAdd(&s_cnt[s], 1.0f); atomicAdd(&s_sum[s], t4.x); atomicAdd(&s_sq[s], t4.x * t4.x);
      }
      {
        int l = min(l4.y, PMAX - 1); lmax = max(lmax, l); int s = l * REP + sub;
        atomicAdd(&s_cnt[s], 1.0f); atomicAdd(&s_sum[s], t4.y); atomicAdd(&s_sq[s], t4.y * t4.y);
      }
      {
        int l = min(l4.z, PMAX - 1); lmax = max(lmax, l); int s = l * REP + sub;
        atomicAdd(&s_cnt[s], 1.0f); atomicAdd(&s_sum[s], t4.z); atomicAdd(&s_sq[s], t4.z * t4.z);
      }
      {
        int l = min(l4.w, PMAX - 1); lmax = max(lmax, l); int s = l * REP + sub;
        atomicAdd(&s_cnt[s], 1.0f); atomicAdd(&s_sum[s], t4.w); atomicAdd(&s_sq[s], t4.w * t4.w);
      }
    }
    // scalar tail
    for (int i = start + n4 + tid; i < end; i += blockDim.x) {
      float t = tags[base + i];
      int l = min(labels[base + i], PMAX - 1);
      lmax = max(lmax, l);
      int s = l * REP + sub;
      atomicAdd(&s_cnt[s], 1.0f); atomicAdd(&s_sum[s], t); atomicAdd(&s_sq[s], t * t);
    }
  }
  atomicMax(&s_maxl, lmax);
  __syncthreads();

  // flush block-local partials to global per-segment scratch
  for (int p = tid; p < P; p += blockDim.x) {
    int q = p * REP;
    float c  = s_cnt[q] + s_cnt[q + 1] + s_cnt[q + 2] + s_cnt[q + 3];
    if (c != 0.0f) {
      float su = s_sum[q] + s_sum[q + 1] + s_sum[q + 2] + s_sum[q + 3];
      float sq = s_sq [q] + s_sq [q + 1] + s_sq [q + 2] + s_sq [q + 3];
      int g = b * P + p;
      atomicAdd(&counts[g], c);
      atomicAdd(&sums[g],   su);
      atomicAdd(&sumsq[g],  sq);
    }
  }
  if (tid == 0 && s_maxl >= 0) atomicMax(&maxlab[b], s_maxl);
}

// ---------------------------------------------------------------------------
// Kernel 2: one wave32 per image. Computes means + pull (Σt² − mean·Σt trick),
// then the P×P push term. The cross term m_p·m_q is a rank-1 outer product:
// computed per 16x16 tile with V_WMMA_F32_16X16X4_F32 (A col / B row in K=0),
// then exp(−(m_p²+m_q²−2·D)) applied elementwise on the WMMA D layout
// (VGPR v, lane l -> M = v + 8*(l>=16), N = l&15).
// ---------------------------------------------------------------------------
__global__ __launch_bounds__(32) void ae_finalize(
    const float* __restrict__ counts, const float* __restrict__ sums,
    const float* __restrict__ sumsq, const int* __restrict__ maxlab,
    const int* __restrict__ pP, int B, float* __restrict__ out)
{
  __shared__ float m[PMAX];
  const int b = blockIdx.x;
  const int lane = threadIdx.x;
  int P = *pP; if (P > PMAX) P = PMAX;

  for (int p = lane; p < PMAX; p += 32) m[p] = 0.0f;
  __syncthreads();

  float pullAcc = 0.0f;
  for (int p = lane; p < P; p += 32) {
    int g = b * P + p;
    float c  = counts[g];
    float su = sums[g];
    float sq = sumsq[g];
    float cm = fmaxf(c, 1.0f);
    float mean = su / cm;
    m[p] = mean;
    // sum (t-mean)^2 over segment == sq - mean*su  (exact when c>0; 0 when empty)
    pullAcc += (sq - mean * su) / cm;
  }
  __syncthreads();
  #pragma unroll
  for (int off = 16; off > 0; off >>= 1) pullAcc += __shfl_xor(pullAcc, off, 32);

  int nt = maxlab[b] + 1; if (nt > P) nt = P;
  const float ntf = (float)nt;

  float acc = 0.0f;
  const int tiles = (nt + 15) >> 4;
  const int l16 = lane & 15;
  const int rhalf = (lane >> 4) << 3;   // 0 for lanes 0-15, 8 for lanes 16-31

  for (int ti = 0; ti < tiles; ++ti) {
    float av = (lane < 16) ? m[ti * 16 + lane] : 0.0f;  // A: 16x4, col K=0 = means rows
    v2f A; A[0] = av; A[1] = 0.0f;
    for (int tj = 0; tj < tiles; ++tj) {
      float bv = (lane < 16) ? m[tj * 16 + lane] : 0.0f; // B: 4x16, row K=0 = means cols
      v2f Bm; Bm[0] = bv; Bm[1] = 0.0f;
      v8f C = {};
      // D[m][n] = m_row[m] * m_col[n]   (rank-1 outer product via WMMA)
      C = __builtin_amdgcn_wmma_f32_16x16x4_f32(false, A, false, Bm,
                                                (short)0, C, false, false);
      const int col = tj * 16 + l16;
      const float mc = m[col];
      const bool cv = (col < nt);
      const int rb = ti * 16 + rhalf;
      #pragma unroll
      for (int v = 0; v < 8; ++v) {
        int row = rb + v;
        float mr = m[row];
        float d2 = mr * mr + mc * mc - 2.0f * C[v];   // (m_row - m_col)^2
        if (cv && row < nt) acc += __expf(-d2);
      }
    }
  }
  #pragma unroll
  for (int off = 16; off > 0; off >>= 1) acc += __shfl_xor(acc, off, 32);

  if (lane == 0) {
    float push = 0.0f;
    if (nt > 1) push = (acc - ntf) / ((ntf - 1.0f) * ntf) * 0.5f;
    float pull = (nt >= 1) ? (pullAcc / fmaxf(ntf, 1.0f)) : 0.0f;
    out[b]     = push;   // push first (reference return order)
    out[B + b] = pull;
  }
}

// ---------------------------------------------------------------------------
extern "C" void kernel_launch(void* const* d_in, const int* in_sizes, int n_in,
                              void* d_out, int out_size, void* d_ws, size_t ws_size,
                              hipStream_t stream) {
  const float* tags   = (const float*)d_in[0];
  const int*   labels = (const int*)d_in[1];
  // d_in[2] (batch_index) is structural: index / NPI — never loaded.
  const int*   pP     = (const int*)d_in[4];   // num_persons (device scalar)

  const int N = in_sizes[0];
  int B = out_size / 2;            // output is [push(B), pull(B)]
  if (B <= 0) B = 1;
  const int NPI = N / B;

  // workspace layout (host-side uses compile-time PMAX so P isn't needed here)
  const int Smax = B * PMAX;
  float* counts = (float*)d_ws;
  float* sums   = counts + Smax;
  float* sumsq  = sums + Smax;
  int*   maxlab = (int*)(sumsq + Smax);

  const int nFloats = 3 * Smax;
  int zgrid = (nFloats + 255) / 256;
  ae_zero_ws<<<zgrid, 256, 0, stream>>>(counts, nFloats, maxlab, B);

  int bpi = 4096 / B; if (bpi < 1) bpi = 1;          // blocks per image
  int perBlock = (NPI + bpi - 1) / bpi;
  perBlock = (perBlock + 3) & ~3;                    // keep float4 alignment
  ae_accum<<<B * bpi, 256, 0, stream>>>(tags, labels, pP,
                                        counts, sums, sumsq, maxlab,
                                        NPI, bpi, perBlock);

  ae_finalize<<<B, 32, 0, stream>>>(counts, sums, sumsq, maxlab, pP, B,
                                    (float*)d_out);
}